// CJL_63471026700598
// MI455X (gfx1250) — compile-verified
//
#include <hip/hip_runtime.h>
#include <hip/hip_bf16.h>

typedef __attribute__((ext_vector_type(2))) float v2f;
typedef __attribute__((ext_vector_type(8))) float v8f;

#define HD 128
__device__ __constant__ float kSlope = (1.0f / 8.0f + 1.0f / 3.0f) * 0.5f;

__device__ __forceinline__ void atom_add_f32(float* p, float v) {
    // lowers to no-return global_atomic_add_f32 (device scope, relaxed)
    __hip_atomic_fetch_add(p, v, __ATOMIC_RELAXED, __HIP_MEMORY_SCOPE_AGENT);
}

// ---------------- Kernel 0: zero workspace (pre_agg + flag) ----------------
__global__ __launch_bounds__(256) void rgcn_zero_kernel(float4* pre4, int* flag,
                                                        int n4, int nflag) {
    int idx = blockIdx.x * blockDim.x + threadIdx.x;
    if (idx < n4) {
        float4 z; z.x = 0.f; z.y = 0.f; z.z = 0.f; z.w = 0.f;
        pre4[idx] = z;
    }
    if (idx < nflag) flag[idx] = 0;
}

// ---------------- Kernel 1: edge gather + pre-projection scatter ----------------
// pre_agg[dst] += h[src] + rel_emb[etype] * time_emb[etime];  flag[dst] = 1
// One wave (32 lanes) per edge; each lane owns 4 consecutive floats (float4).
__global__ __launch_bounds__(256) void rgcn_edge_kernel(
    const float* __restrict__ h, const int* __restrict__ src,
    const int* __restrict__ dst, const int* __restrict__ etype,
    const int* __restrict__ etime, const float* __restrict__ rel,
    const float* __restrict__ timee, float* __restrict__ pre_agg,
    int* __restrict__ flag, int n_edges) {
    int g = blockIdx.x * blockDim.x + threadIdx.x;
    int e = g >> 5;
    int lane = g & 31;
    if (e >= n_edges) return;

    int s  = src[e];
    int d  = dst[e];
    int et = etype[e];
    int tt = etime[e];

    const float4* h4 = reinterpret_cast<const float4*>(h);
    const float4* r4 = reinterpret_cast<const float4*>(rel);
    const float4* t4 = reinterpret_cast<const float4*>(timee);

    float4 hv = h4[(size_t)s  * (HD / 4) + lane];
    float4 rv = r4[(size_t)et * (HD / 4) + lane];
    float4 tv = t4[(size_t)tt * (HD / 4) + lane];

    float m0 = hv.x + rv.x * tv.x;
    float m1 = hv.y + rv.y * tv.y;
    float m2 = hv.z + rv.z * tv.z;
    float m3 = hv.w + rv.w * tv.w;

    float* base = pre_agg + (size_t)d * HD + lane * 4;
    atom_add_f32(base + 0, m0);
    atom_add_f32(base + 1, m1);
    atom_add_f32(base + 2, m2);
    atom_add_f32(base + 3, m3);

    if (lane == 0) flag[d] = 1;  // benign race: all writers store 1
}

// Stage a 128x128 f32 weight matrix into LDS in WMMA B-fragment order.
// Fragment f = k4*8 + n (k4 = K/4 step, n = 16-col tile). Lane layout inside a
// fragment matches V_WMMA_F32_16X16X4_F32 SRC1: lanes 0-15 hold K rows
// {4k4, 4k4+1} for col n*16+lane, lanes 16-31 hold K rows {4k4+2, 4k4+3}.
// Each lane's 2 floats are contiguous -> one conflict-free ds_load_b64.
__device__ __forceinline__ void stage_weight(const float* __restrict__ W,
                                             float* __restrict__ sW, int tid) {
    const float4* W4 = reinterpret_cast<const float4*>(W);
#pragma unroll
    for (int i = 0; i < 16; ++i) {
        int q = tid + i * 256;        // float4 index, 0..4095
        float4 v = W4[q];
        int r    = q >> 5;            // row 0..127
        int c0   = (q & 31) << 2;     // col 0..124 (4-aligned)
        int k4   = r >> 2;
        int rem  = r & 3;
        int half = rem >> 1;          // which lane half owns this K row
        int j    = rem & 1;           // element within the lane's v2f
        int n    = c0 >> 4;
        int lane0 = (half << 4) | (c0 & 15);
        int base = ((k4 * 8 + n) * 32 + lane0) * 2 + j;
        sW[base + 0] = v.x;
        sW[base + 2] = v.y;
        sW[base + 4] = v.z;
        sW[base + 6] = v.w;
    }
}

// ---------------- Kernel 2: fused triple-GEMM via WMMA + rrelu epilogue ----------
// out = leaky( (pre_agg * norm) @ Wn + (h * f) @ Wl + (h * (1-f)) @ We )
// One wave computes a 16-row x 128-col output tile: 8 accumulators of v8f.
// Weights staged per-pass into 64KB LDS; B fragments fetched with ds_load_b64.
__global__ __launch_bounds__(256) void rgcn_gemm_kernel(
    const float* __restrict__ pre_agg, const float* __restrict__ h,
    const float* __restrict__ norm, const int* __restrict__ flag,
    const float* __restrict__ Wn, const float* __restrict__ Wl,
    const float* __restrict__ We, float* __restrict__ out,
    int n_nodes, int n_tiles) {
    __shared__ float sW[HD * HD];  // 64 KB

    int tid  = threadIdx.x;
    int wave = tid >> 5;
    int lane = tid & 31;
    int tile = blockIdx.x * (blockDim.x >> 5) + wave;
    // No early return (all waves must reach the barriers). Out-of-range waves
    // recompute the last tile; their stores duplicate identical values.
    if (tile >= n_tiles) tile = n_tiles - 1;

    int rowBase = tile * 16;
    int mrow    = lane & 15;                 // A row / B col within tile
    int rowA    = rowBase + mrow;
    int rowC    = rowA < n_nodes ? rowA : (n_nodes - 1);  // clamp for loads
    int khalf   = (lane >> 4) << 1;          // lanes 16-31 own K+2,K+3

    v8f acc[8];
#pragma unroll
    for (int n = 0; n < 8; ++n) acc[n] = (v8f)0.0f;

    float nrm = norm[rowC];
    float fl  = flag[rowC] ? 1.0f : 0.0f;

    const v2f* sB = reinterpret_cast<const v2f*>(sW);

#pragma unroll 1
    for (int pass = 0; pass < 3; ++pass) {
        const float* Amat = (pass == 0) ? pre_agg : h;
        const float* Bmat = (pass == 0) ? Wn : (pass == 1 ? Wl : We);
        float scale = (pass == 0) ? nrm : (pass == 1 ? fl : 1.0f - fl);

        __syncthreads();               // WAR: previous pass done reading sW
        stage_weight(Bmat, sW, tid);
        __syncthreads();               // staging visible to all waves

        const float* Arow = Amat + (size_t)rowC * HD;
        for (int k4 = 0; k4 < HD / 4; ++k4) {
            v2f a = *reinterpret_cast<const v2f*>(Arow + k4 * 4 + khalf);
            a.x *= scale;
            a.y *= scale;
#pragma unroll
            for (int n = 0; n < 8; ++n) {
                v2f b = sB[(k4 * 8 + n) * 32 + lane];  // ds_load_b64
                acc[n] = __builtin_amdgcn_wmma_f32_16x16x4_f32(
                    /*neg_a=*/false, a, /*neg_b=*/false, b,
                    /*c_mod=*/(short)0, acc[n],
                    /*reuse_a=*/false, /*reuse_b=*/false);
            }
        }
    }

    // Epilogue: rrelu (eval-mode leaky relu) + guarded store.
    // C/D layout: VGPR j -> row j (lanes 0-15) / row j+8 (lanes 16-31).
    int rowOff = (lane >> 4) << 3;
#pragma unroll
    for (int n = 0; n < 8; ++n) {
#pragma unroll
        for (int j = 0; j < 8; ++j) {
            int row = rowBase + j + rowOff;
            if (row < n_nodes) {
                float x = acc[n][j];
                out[(size_t)row * HD + n * 16 + mrow] = x >= 0.f ? x : x * kSlope;
            }
        }
    }
}

extern "C" void kernel_launch(void* const* d_in, const int* in_sizes, int n_in,
                              void* d_out, int out_size, void* d_ws, size_t ws_size,
                              hipStream_t stream) {
    const float* h     = (const float*)d_in[0];
    const int*   src   = (const int*)d_in[1];
    const int*   dst   = (const int*)d_in[2];
    const int*   etype = (const int*)d_in[3];
    const int*   etime = (const int*)d_in[4];
    const float* norm  = (const float*)d_in[5];
    const float* rel   = (const float*)d_in[6];
    const float* timee = (const float*)d_in[7];
    const float* Wn    = (const float*)d_in[8];
    const float* Wl    = (const float*)d_in[9];
    const float* We    = (const float*)d_in[10];
    float* out = (float*)d_out;

    int n_nodes = in_sizes[0] / HD;
    int n_edges = in_sizes[1];

    float* pre_agg = (float*)d_ws;
    int*   flag    = (int*)((char*)d_ws + (size_t)n_nodes * HD * sizeof(float));

    // 1) zero pre_agg + flag
    int n4 = n_nodes * (HD / 4);
    rgcn_zero_kernel<<<(n4 + 255) / 256, 256, 0, stream>>>(
        (float4*)pre_agg, flag, n4, n_nodes);

    // 2) edge scatter (one wave per edge)
    long long edge_threads = (long long)n_edges * 32;
    rgcn_edge_kernel<<<(int)((edge_threads + 255) / 256), 256, 0, stream>>>(
        h, src, dst, etype, etime, rel, timee, pre_agg, flag, n_edges);

    // 3) fused WMMA GEMMs + epilogue (8 waves/block, one 16x128 tile per wave)
    int n_tiles = (n_nodes + 15) / 16;
    rgcn_gemm_kernel<<<(n_tiles + 7) / 8, 256, 0, stream>>>(
        pre_agg, h, norm, flag, Wn, Wl, We, out, n_nodes, n_tiles);
}